// Baseline_80831284510997
// MI455X (gfx1250) — compile-verified
//
#include <hip/hip_runtime.h>
#include <cmath>

#define Bn   64
#define Cn   100
#define Fn   2048
#define Tn   32
#define Hn   300
#define COMBn 2648
#define HB   (Hn * Bn)       // 19200
#define BOSv 1

typedef __attribute__((ext_vector_type(2))) float v2f;
typedef __attribute__((ext_vector_type(8))) float v8f;

// ---------------------------------------------------------------------------
// meanT[f*64 + b] = mean_c enc[b][c][f]
__global__ void mean_kernel(const float* __restrict__ enc, float* __restrict__ meanT) {
    int tid = blockIdx.x * blockDim.x + threadIdx.x;
    if (tid >= Bn * Fn) return;
    int b = tid / Fn, f = tid % Fn;
    float s = 0.f;
    for (int c = 0; c < Cn; ++c) s += enc[((size_t)(b * Cn + c)) * Fn + f];
    meanT[(size_t)f * Bn + b] = s * (1.0f / Cn);
}

// ---------------------------------------------------------------------------
// e_enc[b*100 + c] = enc[b][c][:] . We_enc    (one wave32 per (b,c))
__global__ void eenc_kernel(const float* __restrict__ enc, const float* __restrict__ We_enc,
                            float* __restrict__ e_enc) {
    int lane = threadIdx.x & 31;
    int wid  = blockIdx.x * (blockDim.x >> 5) + (threadIdx.x >> 5);
    if (wid >= Bn * Cn) return;
    int b = wid / Cn, c = wid % Cn;
    const float* row = enc + ((size_t)(b * Cn + c)) * Fn;
    float v = 0.f;
    for (int f = lane; f < Fn; f += 32) v += row[f] * We_enc[f];
    for (int off = 16; off > 0; off >>= 1) v += __shfl_down(v, off, 32);
    if (lane == 0) e_enc[b * Cn + c] = v;
}

// ---------------------------------------------------------------------------
// embT_all[t][k][b] = embedding[captions[b][t]][k], t in [0, Tn-2]
__global__ void gather_kernel(const float* __restrict__ embedding, const int* __restrict__ caps,
                              float* __restrict__ embT_all) {
    int tid = blockIdx.x * blockDim.x + threadIdx.x;
    if (tid >= (Tn - 1) * HB) return;
    int t = tid / HB, r = tid % HB;
    int k = r / Bn, b = r % Bn;
    int cap = caps[b * Tn + t];
    embT_all[tid] = embedding[(size_t)cap * Hn + k];
}

// out[:, 0, :] = embedding[BOS]
__global__ void pred0_kernel(const float* __restrict__ embedding, float* __restrict__ out) {
    int tid = blockIdx.x * blockDim.x + threadIdx.x;
    if (tid >= Bn * Hn) return;
    int b = tid / Hn, k = tid % Hn;
    out[(size_t)b * (Tn * Hn) + k] = embedding[(size_t)BOSv * Hn + k];
}

// ---------------------------------------------------------------------------
// Generic fp32 WMMA GEMM:  out(64 x N) = AT^T(64 x K) @ W(K x N) + bias [+ extra][tanh]
// AT is K x 64 (transposed activations). One wave per block, 2 N-tiles per wave.
__global__ __launch_bounds__(32) void gemm_wmma_f32(
        const float* __restrict__ AT, const float* __restrict__ W,
        const float* __restrict__ bias, const float* __restrict__ extra,
        float* __restrict__ out, int K, int N, int tout, int ldo, int act) {
    const int lane = threadIdx.x;
    const int half = lane >> 4;          // 0: K pair {k,k+1}, 1: {k+2,k+3}
    const int l    = lane & 15;
    const int m0   = blockIdx.y * 16;
    const int n0a  = blockIdx.x * 32;
    const int n0b  = n0a + 16;
    const int m    = m0 + l;
    const int na   = n0a + l;
    const int nb   = n0b + l;
    const int nca  = na < N ? na : N - 1;   // clamp so no OOB, keep EXEC all-ones
    const int ncb  = nb < N ? nb : N - 1;

    const float* Ap  = AT + (size_t)(2 * half) * Bn + m;
    const float* Wpa = W  + (size_t)(2 * half) * N + nca;
    const float* Wpb = W  + (size_t)(2 * half) * N + ncb;

    v8f acc0 = {};
    v8f acc1 = {};
    for (int k = 0; k < K; k += 4) {
        v2f a, b0, b1;
        a.x  = Ap[(size_t)k * Bn];
        a.y  = Ap[(size_t)(k + 1) * Bn];
        b0.x = Wpa[(size_t)k * N];
        b0.y = Wpa[(size_t)(k + 1) * N];
        b1.x = Wpb[(size_t)k * N];
        b1.y = Wpb[(size_t)(k + 1) * N];
        acc0 = __builtin_amdgcn_wmma_f32_16x16x4_f32(false, a, false, b0, (short)0, acc0, false, false);
        acc1 = __builtin_amdgcn_wmma_f32_16x16x4_f32(false, a, false, b1, (short)0, acc1, false, false);
    }

#pragma unroll
    for (int r = 0; r < 8; ++r) {
        int mo = m0 + r + 8 * half;
        if (na < N) {
            float v = acc0[r] + bias[na];
            size_t idx = tout ? ((size_t)na * Bn + mo) : ((size_t)mo * ldo + na);
            if (extra) v += extra[idx];
            if (act) v = tanhf(v);
            out[idx] = v;
        }
        if (nb < N) {
            float v = acc1[r] + bias[nb];
            size_t idx = tout ? ((size_t)nb * Bn + mo) : ((size_t)mo * ldo + nb);
            if (extra) v += extra[idx];
            if (act) v = tanhf(v);
            out[idx] = v;
        }
    }
}

// ---------------------------------------------------------------------------
// Per-step attention: copies emb_t into combT[0:300), reads h from combT[300:600),
// softmax over C, writes ctx into combT[600:2648). One block per batch row.
__global__ __launch_bounds__(256) void attn_kernel(
        const float* __restrict__ enc, const float* __restrict__ e_enc,
        const float* __restrict__ We_hid, const float* __restrict__ be,
        const float* __restrict__ embT_t, float* __restrict__ combT) {
    const int b = blockIdx.x;
    const int tid = threadIdx.x;
    __shared__ float red[256];
    __shared__ float attn[Cn];
    __shared__ float sc[2];

    // copy embedding rows for this step
    for (int k = tid; k < Hn; k += 256)
        combT[(size_t)k * Bn + b] = embT_t[(size_t)k * Bn + b];

    // eh = h . We_hid  (h lives in combT rows 300..599)
    float p = 0.f;
    for (int j = tid; j < Hn; j += 256)
        p += combT[(size_t)(Hn + j) * Bn + b] * We_hid[j];
    red[tid] = p;
    __syncthreads();
    for (int s = 128; s > 0; s >>= 1) {
        if (tid < s) red[tid] += red[tid + s];
        __syncthreads();
    }
    const float eh = red[0];

    if (tid < Cn) attn[tid] = e_enc[b * Cn + tid] + eh + *be;
    __syncthreads();
    if (tid == 0) {
        float mx = attn[0];
        for (int c = 1; c < Cn; ++c) mx = fmaxf(mx, attn[c]);
        sc[0] = mx;
    }
    __syncthreads();
    if (tid < Cn) attn[tid] = expf(attn[tid] - sc[0]);
    __syncthreads();
    if (tid == 0) {
        float s = 0.f;
        for (int c = 0; c < Cn; ++c) s += attn[c];
        sc[1] = 1.0f / s;
    }
    __syncthreads();
    if (tid < Cn) attn[tid] *= sc[1];
    __syncthreads();

    // ctx[f] = sum_c attn[c] * enc[b][c][f]  -> combT rows 600..2647
    const float* eb = enc + (size_t)b * Cn * Fn;
    for (int f = tid; f < Fn; f += 256) {
        float a = 0.f;
        for (int c = 0; c < Cn; ++c) a += attn[c] * eb[(size_t)c * Fn + f];
        combT[(size_t)(2 * Hn + f) * Bn + b] = a;
    }
}

// ---------------------------------------------------------------------------
// LSTM pointwise update: gatesT = [i|f|o|g] pre-activations, each (Hn x Bn) transposed.
// Writes new cT and new h into combT rows 300..599.
__global__ void lstm_update_kernel(const float* __restrict__ gatesT,
                                   float* __restrict__ cT, float* __restrict__ combT) {
    int idx = blockIdx.x * blockDim.x + threadIdx.x;
    if (idx >= HB) return;
    float ip = gatesT[idx];
    float fp = gatesT[HB + idx];
    float op = gatesT[2 * HB + idx];
    float gp = gatesT[3 * HB + idx];
    float i = 1.0f / (1.0f + expf(-ip));
    float f = 1.0f / (1.0f + expf(-fp));
    float o = 1.0f / (1.0f + expf(-op));
    float g = tanhf(gp);
    float cn = f * cT[idx] + i * g;
    cT[idx] = cn;
    combT[HB + idx] = o * tanhf(cn);   // h region starts at row 300 => offset 300*64
}

// ---------------------------------------------------------------------------
extern "C" void kernel_launch(void* const* d_in, const int* in_sizes, int n_in,
                              void* d_out, int out_size, void* d_ws, size_t ws_size,
                              hipStream_t stream) {
    const float* enc    = (const float*)d_in[0];
    const int*   caps   = (const int*)d_in[1];
    const float* emb    = (const float*)d_in[2];
    const float* Wh0    = (const float*)d_in[3];
    const float* bh0    = (const float*)d_in[4];
    const float* Wc0    = (const float*)d_in[5];
    const float* bc0    = (const float*)d_in[6];
    const float* We_enc = (const float*)d_in[7];
    const float* We_hid = (const float*)d_in[8];
    const float* be     = (const float*)d_in[9];
    const float* Wi     = (const float*)d_in[10];
    const float* bi     = (const float*)d_in[11];
    const float* Wf     = (const float*)d_in[12];
    const float* bf     = (const float*)d_in[13];
    const float* Wo     = (const float*)d_in[14];
    const float* bo     = (const float*)d_in[15];
    const float* Wg     = (const float*)d_in[16];
    const float* bg     = (const float*)d_in[17];
    const float* Wcp    = (const float*)d_in[18];
    const float* bcp    = (const float*)d_in[19];
    const float* Whp    = (const float*)d_in[20];
    const float* bhp    = (const float*)d_in[21];
    const float* Wop    = (const float*)d_in[22];
    const float* bop    = (const float*)d_in[23];
    float* out = (float*)d_out;
    float* ws  = (float*)d_ws;

    size_t o = 0;
    float* meanT  = ws + o; o += (size_t)Fn * Bn;        // 2048x64
    float* e_enc  = ws + o; o += (size_t)Bn * Cn;        // 64x100
    float* combT  = ws + o; o += (size_t)COMBn * Bn;     // 2648x64
    float* cT     = ws + o; o += HB;                     // 300x64
    float* gatesT = ws + o; o += 4 * (size_t)HB;         // 4x300x64
    float* ctxpT  = ws + o; o += HB;
    float* sT     = ws + o; o += HB;
    float* embT   = ws + o; o += (size_t)(Tn - 1) * HB;  // 31x300x64

    const dim3 ggrid((Hn + 31) / 32, Bn / 16);           // (10, 4) waves of 32
    const dim3 gblk(32);

    // ---- setup ----
    mean_kernel<<<(Bn * Fn + 255) / 256, 256, 0, stream>>>(enc, meanT);
    eenc_kernel<<<(Bn * Cn + 7) / 8, 256, 0, stream>>>(enc, We_enc, e_enc);
    gather_kernel<<<((Tn - 1) * HB + 255) / 256, 256, 0, stream>>>(emb, caps, embT);
    pred0_kernel<<<(Bn * Hn + 255) / 256, 256, 0, stream>>>(emb, out);

    // h0 -> combT rows 300..599 (transposed, tanh); c0 -> cT
    gemm_wmma_f32<<<ggrid, gblk, 0, stream>>>(meanT, Wh0, bh0, nullptr, combT + HB, Fn, Hn, 1, 0, 1);
    gemm_wmma_f32<<<ggrid, gblk, 0, stream>>>(meanT, Wc0, bc0, nullptr, cT, Fn, Hn, 1, 0, 1);

    // ---- recurrence ----
    for (int t = 0; t < Tn - 1; ++t) {
        const float* embT_t = embT + (size_t)t * HB;

        attn_kernel<<<Bn, 256, 0, stream>>>(enc, e_enc, We_hid, be, embT_t, combT);

        gemm_wmma_f32<<<ggrid, gblk, 0, stream>>>(combT, Wi, bi, nullptr, gatesT + 0 * HB, COMBn, Hn, 1, 0, 0);
        gemm_wmma_f32<<<ggrid, gblk, 0, stream>>>(combT, Wf, bf, nullptr, gatesT + 1 * HB, COMBn, Hn, 1, 0, 0);
        gemm_wmma_f32<<<ggrid, gblk, 0, stream>>>(combT, Wo, bo, nullptr, gatesT + 2 * HB, COMBn, Hn, 1, 0, 0);
        gemm_wmma_f32<<<ggrid, gblk, 0, stream>>>(combT, Wg, bg, nullptr, gatesT + 3 * HB, COMBn, Hn, 1, 0, 0);

        // ctxpT = ctx @ Wcp + bcp + emb_t   (ctx lives in combT rows 600..2647)
        gemm_wmma_f32<<<ggrid, gblk, 0, stream>>>(combT + 2 * HB, Wcp, bcp, embT_t, ctxpT, Fn, Hn, 1, 0, 0);

        lstm_update_kernel<<<(HB + 255) / 256, 256, 0, stream>>>(gatesT, cT, combT);

        // sT = h_new @ Whp + bhp + ctxpT
        gemm_wmma_f32<<<ggrid, gblk, 0, stream>>>(combT + HB, Whp, bhp, ctxpT, sT, Hn, Hn, 1, 0, 0);

        // pred -> out[:, t+1, :]  (row-major, row stride Tn*Hn)
        gemm_wmma_f32<<<ggrid, gblk, 0, stream>>>(sT, Wop, bop, nullptr,
                                                  out + (size_t)(t + 1) * Hn, Hn, Hn, 0, Tn * Hn, 0);
    }
}